// MeanAggregator_8065948582555
// MI455X (gfx1250) — compile-verified
//
#include <hip/hip_runtime.h>
#include <stdint.h>

// ---------------------------------------------------------------------------
// MeanAggregator: out[b,:] = mean_k features[neigh_idx[b,k], :]
//   features : [V=100000, 128] f32   (51.2 MB -> L2-resident on MI455X)
//   neigh_idx: [B=50000, 10]  int32
//   out      : [B, 128] f32
// Memory-bound gather-mean (256 MB gathered reads, mostly L2 hits; 64 MFLOP).
// Strategy: 1 wave per row; async-gather 10 x 512B rows into LDS
// (global_load_async_to_lds_b128, ASYNCcnt), reduce with ds_load_b128 +
// v_add_f32, non-temporal b128 store of the result.
// ---------------------------------------------------------------------------

#define D_DIM 128
#define K_NBR 10
#define WAVES_PER_BLOCK 8
#define BLOCK_THREADS (WAVES_PER_BLOCK * 32)

typedef float f32x4 __attribute__((ext_vector_type(4)));
typedef int   v4i   __attribute__((ext_vector_type(4)));
typedef __attribute__((address_space(1))) v4i* g_v4i_ptr;   // global b128 src
typedef __attribute__((address_space(3))) v4i* l_v4i_ptr;   // LDS b128 dst
typedef __attribute__((address_space(3))) const f32x4* lds_f32x4_cptr;

__device__ __forceinline__ void async_gather_b128(const void* gsrc, uint32_t lds_off) {
#if __has_builtin(__builtin_amdgcn_global_load_async_to_lds_b128)
  __builtin_amdgcn_global_load_async_to_lds_b128(
      (g_v4i_ptr)(uintptr_t)gsrc, (l_v4i_ptr)lds_off, /*offset=*/0, /*cpol=*/0);
#else
  // GV mode: VDST = LDS byte address, VADDR = 64-bit global address
  asm volatile("global_load_async_to_lds_b128 %0, %1, off"
               :
               : "v"(lds_off), "v"(gsrc)
               : "memory");
#endif
}

__device__ __forceinline__ void wait_async0() {
#if __has_builtin(__builtin_amdgcn_s_wait_asynccnt)
  __builtin_amdgcn_s_wait_asynccnt(0);
#else
  asm volatile("s_wait_asynccnt 0" ::: "memory");
#endif
}

__global__ __launch_bounds__(BLOCK_THREADS) void MeanAggregator_kernel(
    const float* __restrict__ features,
    const int*   __restrict__ neigh_idx,
    float* __restrict__ out,
    int n_rows)
{
  // 8 waves x 10 rows x 512B = 40 KB (WGP has 320 KB -> ~8 blocks resident)
  __shared__ float tile[WAVES_PER_BLOCK][K_NBR][D_DIM];

  const int lane = threadIdx.x & 31;
  const int wave = threadIdx.x >> 5;
  const int row  = blockIdx.x * WAVES_PER_BLOCK + wave;
  if (row >= n_rows) return;   // wave-uniform

  // Indices streamed once -> non-temporal load; lanes 0..9 each grab one.
  int my_idx = 0;
  if (lane < K_NBR)
    my_idx = __builtin_nontemporal_load(&neigh_idx[(size_t)row * K_NBR + lane]);

  const uint32_t lane_byte = (uint32_t)lane * 16u;                 // float4 per lane
  uint32_t lds_base = (uint32_t)(uintptr_t)&tile[wave][0][0];      // LDS byte offset

  // Issue 10 async row gathers (each instruction moves one 512B feature row).
#pragma unroll
  for (int k = 0; k < K_NBR; ++k) {
    const int ridx = __builtin_amdgcn_readlane(my_idx, k);
    const char* gsrc =
        (const char*)(features + (size_t)ridx * D_DIM) + lane_byte;
    async_gather_b128(gsrc, lds_base + (uint32_t)k * (D_DIM * 4u) + lane_byte);
  }
  wait_async0();

  // Launder the LDS offset so the compiler cannot assume the tile is
  // unwritten; keep the addrspace(3) type so reads lower to ds_load_b128.
  asm volatile("" : "+v"(lds_base) :: "memory");

  f32x4 acc = {0.f, 0.f, 0.f, 0.f};
#pragma unroll
  for (int k = 0; k < K_NBR; ++k) {
    lds_f32x4_cptr p =
        (lds_f32x4_cptr)(lds_base + (uint32_t)k * (D_DIM * 4u) + lane_byte);
    acc += *p;
  }
  acc *= (1.0f / (float)K_NBR);

  // Output is write-once: non-temporal b128 store, keep L2 for the table.
  __builtin_nontemporal_store(acc, (f32x4*)(out + (size_t)row * D_DIM) + lane);
}

extern "C" void kernel_launch(void* const* d_in, const int* in_sizes, int n_in,
                              void* d_out, int out_size, void* d_ws, size_t ws_size,
                              hipStream_t stream) {
  const float* features  = (const float*)d_in[0];
  const int*   neigh_idx = (const int*)d_in[1];
  float*       out       = (float*)d_out;

  const int n_rows = in_sizes[1] / K_NBR;            // B = (B*K)/K
  const int grid   = (n_rows + WAVES_PER_BLOCK - 1) / WAVES_PER_BLOCK;

  MeanAggregator_kernel<<<grid, BLOCK_THREADS, 0, stream>>>(
      features, neigh_idx, out, n_rows);
}